// SafetyConstraint_68238440398980
// MI455X (gfx1250) — compile-verified
//
#include <hip/hip_runtime.h>
#include <cstdint>
#include <cstddef>

// ---------------------------------------------------------------------------
// CBF safety-constraint kernel for MI455X (gfx1250).
// Memory-bound elementwise row kernel: 92 B I/O per row, ~250 flops per row.
// Data path: CDNA5 async global->LDS copies (ASYNCcnt) with linear b128
// coalescing, s_wait_asynccnt + workgroup barrier, conflict-free LDS reads.
// ---------------------------------------------------------------------------

#define CBF_EPS   1e-8f
#define CBF_BETA  5.0f

__device__ __forceinline__ void async_copy_b128(uint32_t lds_byte_addr, const void* gaddr) {
  // VDST = LDS byte address VGPR, VADDR = 64-bit global address VGPR pair.
  asm volatile("global_load_async_to_lds_b128 %0, %1, off"
               :: "v"(lds_byte_addr), "v"(gaddr)
               : "memory");
}

__device__ __forceinline__ void wait_asynccnt0() {
  asm volatile("s_wait_asynccnt 0" ::: "memory");
}

__global__ __launch_bounds__(256) void cbf_kernel(const float* __restrict__ state,
                                                  const float* __restrict__ action,
                                                  float* __restrict__ out,
                                                  long long nrows) {
  __shared__ __align__(16) float sTile[256 * 18];  // 18 KB
  __shared__ __align__(16) float aTile[256 * 4];   //  4 KB

  const int       tid  = threadIdx.x;
  const long long row0 = (long long)blockIdx.x * 256;
  const long long row  = row0 + tid;
  const long long rem  = nrows - row0;
  const bool full      = (rem >= 256);

  float s[18], act[4];

  if (full) {
    // Stage 256 rows of state (1152 x 16B chunks) + action (256 x 16B chunks)
    // into LDS with linear, perfectly-coalesced async b128 copies.
    const char* gs = (const char*)(state + row0 * 18);
    const char* ga = (const char*)(action + row0 * 4);
    const uint32_t sbase = (uint32_t)(uintptr_t)&sTile[0];
    const uint32_t abase = (uint32_t)(uintptr_t)&aTile[0];

    for (int k = tid; k < 1152; k += 256) {
      async_copy_b128(sbase + (uint32_t)(k << 4), gs + ((size_t)k << 4));
    }
    async_copy_b128(abase + (uint32_t)(tid << 4), ga + ((size_t)tid << 4));

    wait_asynccnt0();   // this wave's async copies landed in LDS
    __syncthreads();    // all waves' copies landed

#pragma unroll
    for (int j = 0; j < 18; ++j) s[j] = sTile[tid * 18 + j];
#pragma unroll
    for (int j = 0; j < 4; ++j) act[j] = aTile[tid * 4 + j];
  } else {
    if (row < nrows) {
      const float* sp = state + row * 18;
#pragma unroll
      for (int j = 0; j < 18; ++j) s[j] = sp[j];
      const float* ap = action + row * 4;
#pragma unroll
      for (int j = 0; j < 4; ++j) act[j] = ap[j];
    }
  }

  if (row >= nrows) return;

  // ---- unpack ----
  const float px = s[0],  py = s[1],  pz = s[2];
  const float q0 = s[3],  q1 = s[4],  q2 = s[5],  q3 = s[6];
  const float vx = s[7],  vy = s[8],  vz = s[9];
  const float ox = s[10], oy = s[11], oz = s[12];
  // s[13..16] = controls (zero gradient path), s[17] = target (sdot = 0)

  // ---- normalized quaternion ----
  const float nq   = sqrtf(q0 * q0 + q1 * q1 + q2 * q2 + q3 * q3);
  const float dq   = nq + CBF_EPS;
  const float invd = 1.0f / dq;
  const float qw = q0 * invd, qx = q1 * invd, qy = q2 * invd, qz = q3 * invd;

  // ---- barriers ----
  const float b0 = pz - 0.4f;                              // alt low
  const float b1 = 1.6f - pz;                              // alt high
  const float b2 = 1.0f - px * px;                         // pos x
  const float b3 = 1.0f - py * py;                         // pos y
  const float b4 = 0.16f - (vx * vx + vy * vy + vz * vz);  // vel
  const float aq  = fabsf(qw);
  const float qwc = fminf(fmaxf(aq, 0.1f), 1.0f);
  const float b5  = qwc - 0.75f;                           // tilt

  // ---- smooth-min via stable logsumexp; softmax weights = dh/db_j ----
  const float z0 = -CBF_BETA * b0, z1 = -CBF_BETA * b1, z2 = -CBF_BETA * b2;
  const float z3 = -CBF_BETA * b3, z4 = -CBF_BETA * b4, z5 = -CBF_BETA * b5;
  const float m  = fmaxf(fmaxf(fmaxf(z0, z1), fmaxf(z2, z3)), fmaxf(z4, z5));
  const float e0 = expf(z0 - m), e1 = expf(z1 - m), e2 = expf(z2 - m);
  const float e3 = expf(z3 - m), e4 = expf(z4 - m), e5 = expf(z5 - m);
  const float S    = e0 + e1 + e2 + e3 + e4 + e5;
  const float h    = -(m + logf(S)) * (1.0f / CBF_BETA);
  const float invS = 1.0f / S;
  const float w0 = e0 * invS, w1 = e1 * invS, w2 = e2 * invS;
  const float w3 = e3 * invS, w4 = e4 * invS, w5 = e5 * invS;

  // ---- lie = <grad_h(state), sdot> ; only pos/quat/vel_z terms survive ----
  // pos grads dotted with sdot[0:3] = vel
  float lie = (-2.0f * w2 * px) * vx + (-2.0f * w3 * py) * vy + (w0 - w1) * vz;

  // quat grads dotted with quat_dot
  const float qd0 = 0.5f * (-qx * ox - qy * oy - qz * oz);
  const float qd1 = 0.5f * ( qw * ox + qy * oz - qz * oy);
  const float qd2 = 0.5f * ( qw * oy + qz * ox - qx * oz);
  const float qd3 = 0.5f * ( qw * oz + qx * oy - qy * ox);
  const float sgn  = (q0 > 0.0f) ? 1.0f : ((q0 < 0.0f) ? -1.0f : 0.0f);
  const float mask = (aq >= 0.1f && aq <= 1.0f) ? 1.0f : 0.0f;  // clip VJP
  const float qdot = q0 * qd0 + q1 * qd1 + q2 * qd2 + q3 * qd3;
  // d|qw|/dq_k = sgn * (delta_k0/d - q0*q_k/(n*d^2))
  lie += (w5 * mask * sgn) * (qd0 * invd - q0 * qdot * invd * invd / nq);

  // vel grad z dotted with vel_dot z
  const float thrust = fminf(fmaxf(act[0], 0.0f), 0.35f);
  const float R33    = fminf(fmaxf(1.0f - 2.0f * (qx * qx + qy * qy), -1.0f), 1.0f);
  const float az     = thrust * R33 * (1.0f / 0.027f) - 9.81f;
  lie += (-2.0f * w4 * vz) * az;

  // ---- clip chain ----
  const float hdot = fminf(fmaxf(lie, -100.0f), 100.0f);
  const float cbf  = fminf(fmaxf(hdot + 2.0f * h, -20.0f), 20.0f);
  out[row] = cbf;
}

extern "C" void kernel_launch(void* const* d_in, const int* in_sizes, int n_in,
                              void* d_out, int out_size, void* d_ws, size_t ws_size,
                              hipStream_t stream) {
  (void)n_in; (void)d_ws; (void)ws_size; (void)out_size;
  const float* state  = (const float*)d_in[0];
  const float* action = (const float*)d_in[1];
  float*       out    = (float*)d_out;

  const long long nrows  = (long long)in_sizes[0] / 18;
  const int       blocks = (int)((nrows + 255) / 256);

  cbf_kernel<<<blocks, 256, 0, stream>>>(state, action, out, nrows);
}